// Phi_Layer_joint_opt_32212254720395
// MI455X (gfx1250) — compile-verified
//
#include <hip/hip_runtime.h>
#include <math.h>

// Problem constants (from reference)
#define NSC  32
#define KDIM 4096   // Nt*Nr  (GEMM reduction dim)
#define QDIM 1024   // Mt*Mr  (GEMM M dim)
#define GSQ  4096   // G*G    (Phi N dim)
#define YCOLS 2048  // B*NSC  (y N dim, batched into one GEMM)
#define KB   16     // K-depth per LDS stage

typedef float v2f __attribute__((ext_vector_type(2)));
typedef float v8f __attribute__((ext_vector_type(8)));
typedef int   v4i __attribute__((ext_vector_type(4)));

// ---------------------------------------------------------------------------
// Async global->LDS copy (CDNA5): gated so absence of the builtin still builds.
// Probe-discovered signature: (v4i __device__* src, v4i __shared__* dst, imm, imm)
// ---------------------------------------------------------------------------
#if __has_builtin(__builtin_amdgcn_global_load_async_to_lds_b128)
#define HAVE_ASYNC_LDS 1
#else
#define HAVE_ASYNC_LDS 0
#endif

typedef __attribute__((address_space(1))) v4i gv4i_t;
typedef __attribute__((address_space(3))) v4i lv4i_t;

__device__ __forceinline__ void copy16(const float* __restrict__ g, void* l) {
#if HAVE_ASYNC_LDS
    __builtin_amdgcn_global_load_async_to_lds_b128(
        (gv4i_t*)g,
        (lv4i_t*)l,
        0, 0);
#else
    *(float4*)l = *(const float4*)g;
#endif
}

__device__ __forceinline__ void wait_stage() {
#if HAVE_ASYNC_LDS
#if __has_builtin(__builtin_amdgcn_s_wait_asynccnt)
    __builtin_amdgcn_s_wait_asynccnt(0);
#else
    asm volatile("s_wait_asynccnt 0x0" ::: "memory");
#endif
#endif
    __syncthreads();
}

// ---------------------------------------------------------------------------
// Stage 0: phase tables  Wr|Wi|Fr|Fi  (each 2048 floats), scaled by 1/sqrt(64)
// ---------------------------------------------------------------------------
__global__ void phases_kernel(const float* __restrict__ kW,
                              const float* __restrict__ kF,
                              float* __restrict__ wf) {
    int i = blockIdx.x * blockDim.x + threadIdx.x;
    if (i < 2048) {
        float s, c;
        __sincosf(kW[i], &s, &c);
        wf[i]        = c * 0.125f;   // Wr
        wf[2048 + i] = s * 0.125f;   // Wi
        __sincosf(kF[i], &s, &c);
        wf[4096 + i] = c * 0.125f;   // Fr
        wf[6144 + i] = s * 0.125f;   // Fi
    }
}

// ---------------------------------------------------------------------------
// Stage 1: Qc[q, n] = F[nt,mt] * conj(W[nr,mr]), interleaved (q, n, 2)
// ---------------------------------------------------------------------------
__global__ void build_q_kernel(const float* __restrict__ wf,
                               float* __restrict__ Qc) {
    int n = blockIdx.x * blockDim.x + threadIdx.x;   // 0..4095
    int q = blockIdx.y;                              // 0..1023
    int mt = q >> 5, mr = q & 31;
    int nt = n >> 6, nr = n & 63;
    float fr = wf[4096 + nt * 32 + mt];
    float fi = wf[6144 + nt * 32 + mt];
    float wr = wf[nr * 32 + mr];
    float wi = wf[2048 + nr * 32 + mr];
    v2f qc;
    qc.x = fr * wr + fi * wi;        // f * conj(w)
    qc.y = fi * wr - fr * wi;
    ((v2f*)Qc)[(size_t)q * KDIM + n] = qc;
}

// ---------------------------------------------------------------------------
// Stage 2: pack h:  hc[n, col] = H[b, nr, nt, s],  col = b*32+s, n = nt*64+nr
// ---------------------------------------------------------------------------
__global__ void pack_h_kernel(const float* __restrict__ H,
                              float* __restrict__ hc) {
    int col = blockIdx.x * blockDim.x + threadIdx.x; // 0..2047
    int n   = blockIdx.y;                            // 0..4095
    int b = col >> 5, s = col & 31;
    int nt = n >> 6, nr = n & 63;
    v2f v = ((const v2f*)H)[(((size_t)b * 64 + nr) * 64 + nt) * 32 + s];
    ((v2f*)hc)[(size_t)n * YCOLS + col] = v;
}

// ---------------------------------------------------------------------------
// Stage 3: effective noise  effc[q, col] = sum_nr conj(W[nr,mr]) *
//          noise[b, mr>>2, nr, mt*32+s]      (q = mt*32+mr, col = b*32+s)
// ---------------------------------------------------------------------------
__global__ void eff_kernel(const float* __restrict__ noise,
                           const float* __restrict__ wf,
                           float* __restrict__ effc) {
    int col = blockIdx.x * blockDim.x + threadIdx.x; // 0..2047
    int q   = blockIdx.y;                            // 0..1023
    int mt = q >> 5, mr = q & 31;
    int r  = mr >> 2;
    int b = col >> 5, s = col & 31;
    const v2f* nz = (const v2f*)noise;
    size_t base = (((size_t)b * 8 + r) * 64) * 1024 + (size_t)(mt * 32 + s);
    float ar = 0.f, ai = 0.f;
    #pragma unroll 4
    for (int nr = 0; nr < 64; ++nr) {
        float wr = wf[nr * 32 + mr];
        float wi = wf[2048 + nr * 32 + mr];
        v2f nv = nz[base + (size_t)nr * 1024];
        ar += wr * nv.x + wi * nv.y;      // conj(w) * n
        ai += wr * nv.y - wi * nv.x;
    }
    ((v2f*)effc)[(size_t)q * YCOLS + col] = (v2f){ar, ai};
}

// ---------------------------------------------------------------------------
// Complex GEMM via V_WMMA_F32_16X16X4_F32 with double-buffered async LDS
// staging; all fragments of a stage are fetched up-front so the 64 WMMAs of
// the stage run behind a single DS-count wait.
//   out(M=1024, N=ldb) = A(1024, 4096) * B(4096, ldb)   (interleaved complex)
// Workgroup 64(M) x 128(N), 8 waves, wave tile 32x32 (2x2 sub-tiles).
// A LDS: K-pair float4s, row stride 9 (bank-conflict free b128 fragment reads)
// B LDS: (k, col) complex, row stride 130 (conflict-free b64 fragment reads)
// MODE 0: plain interleaved store (Phi). MODE 1: += effc, scatter to y layout.
// ---------------------------------------------------------------------------
template <int MODE>
__global__ __launch_bounds__(256) void cgemm_wmma_kernel(
    const float* __restrict__ A,     // Qc  (QDIM x KDIM x 2)
    const float* __restrict__ Bm,    // (KDIM x ldb x 2)
    const float* __restrict__ effc,  // (QDIM x YCOLS x 2), MODE==1 only
    float* __restrict__ out,
    int ldb) {
    __shared__ float4 aLds[2][64 * 9];     //  9.0 KB per buffer
    __shared__ v2f    bLds[2][KB * 130];   // 16.3 KB per buffer

    const int tid  = threadIdx.x;
    const int lane = tid & 31;
    const int wid  = tid >> 5;
    const int mw = wid >> 2;                 // 0..1
    const int nw = wid & 3;                  // 0..3
    const int hl  = lane >> 4;               // lane half: 0/1
    const int l15 = lane & 15;
    const size_t m_blk = (size_t)blockIdx.y * 64;
    const size_t n_blk = (size_t)blockIdx.x * 128;

    v8f accR[2][2] = {};
    v8f accI[2][2] = {};

    // Cooperative stage loader: A = 512 x b128 chunks, B = 1024 x b128 chunks.
    auto load_stage = [&](int kb, int bf) {
        const int k0 = kb * KB;
        #pragma unroll
        for (int t = 0; t < 2; ++t) {            // A tile: 64 rows x 8 K-pairs
            int idx = tid + t * 256;
            int row = idx >> 3, j = idx & 7;
            const float* g = A + ((m_blk + row) * KDIM + (size_t)(k0 + j * 2)) * 2;
            copy16(g, &aLds[bf][row * 9 + j]);
        }
        #pragma unroll
        for (int t = 0; t < 4; ++t) {            // B tile: 16 K x 64 col-pairs
            int idx = tid + t * 256;
            int k = idx >> 6, cp = idx & 63;
            const float* g = Bm + ((size_t)(k0 + k) * ldb + n_blk + cp * 2) * 2;
            copy16(g, &bLds[bf][k * 130 + cp * 2]);
        }
    };

    load_stage(0, 0);
    wait_stage();

    const int NB = KDIM / KB;                    // 256 stages
    for (int kb = 0; kb < NB; ++kb) {
        const int bf = kb & 1;
        if (kb + 1 < NB) load_stage(kb + 1, bf ^ 1);   // async prefetch stage

        // Fetch ALL raw fragments of this stage first (24 DS loads, distinct
        // registers) so the WMMA block below runs behind one dscnt wait.
        float4 fa[4][2];
        v2f p0[4][2], p1[4][2];
        #pragma unroll
        for (int ks = 0; ks < 4; ++ks) {
            const int jj = ks * 2 + hl;          // K-pair index for this lane
            #pragma unroll
            for (int m = 0; m < 2; ++m)
                fa[ks][m] = aLds[bf][(mw * 32 + m * 16 + l15) * 9 + jj];
            const int kk = ks * 4 + hl * 2;
            #pragma unroll
            for (int n = 0; n < 2; ++n) {
                int c = nw * 32 + n * 16 + l15;
                p0[ks][n] = bLds[bf][kk * 130 + c];
                p1[ks][n] = bLds[bf][(kk + 1) * 130 + c];
            }
        }

        #pragma unroll
        for (int ks = 0; ks < 4; ++ks) {         // 4 WMMA K-steps per stage
            v2f ar[2], ai[2], ain[2], br[2], bi[2];
            #pragma unroll
            for (int m = 0; m < 2; ++m) {
                ar[m]  = (v2f){fa[ks][m].x, fa[ks][m].z};
                ai[m]  = (v2f){fa[ks][m].y, fa[ks][m].w};
                ain[m] = -ai[m];                 // f32 WMMA has no A-neg bit
            }
            #pragma unroll
            for (int n = 0; n < 2; ++n) {
                br[n] = (v2f){p0[ks][n].x, p1[ks][n].x};
                bi[n] = (v2f){p0[ks][n].y, p1[ks][n].y};
            }
            #pragma unroll
            for (int m = 0; m < 2; ++m)
                #pragma unroll
                for (int n = 0; n < 2; ++n) {
                    accR[m][n] = __builtin_amdgcn_wmma_f32_16x16x4_f32(
                        false, ar[m], false, br[n], (short)0, accR[m][n], false, false);
                    accR[m][n] = __builtin_amdgcn_wmma_f32_16x16x4_f32(
                        false, ain[m], false, bi[n], (short)0, accR[m][n], false, false);
                    accI[m][n] = __builtin_amdgcn_wmma_f32_16x16x4_f32(
                        false, ar[m], false, bi[n], (short)0, accI[m][n], false, false);
                    accI[m][n] = __builtin_amdgcn_wmma_f32_16x16x4_f32(
                        false, ai[m], false, br[n], (short)0, accI[m][n], false, false);
                }
        }
        wait_stage();   // next-stage data landed in LDS; all waves done reading
    }

    // Epilogue. C/D layout: VGPR v, lanes 0-15 -> M=v, lanes 16-31 -> M=v+8.
    #pragma unroll
    for (int m = 0; m < 2; ++m)
        #pragma unroll
        for (int n = 0; n < 2; ++n) {
            int colc = (int)n_blk + nw * 32 + n * 16 + l15;
            #pragma unroll
            for (int v = 0; v < 8; ++v) {
                int row = (int)m_blk + mw * 32 + m * 16 + v + hl * 8;
                v2f val = (v2f){accR[m][n][v], accI[m][n][v]};
                if (MODE == 0) {
                    ((v2f*)out)[(size_t)row * ldb + colc] = val;
                } else {
                    v2f e = ((const v2f*)effc)[(size_t)row * YCOLS + colc];
                    val.x += e.x;
                    val.y += e.y;
                    int b = colc >> 5, s = colc & 31;
                    ((v2f*)out)[((size_t)(b * QDIM + row)) * NSC + s] = val;
                }
            }
        }
}

// ---------------------------------------------------------------------------
extern "C" void kernel_launch(void* const* d_in, const int* in_sizes, int n_in,
                              void* d_out, int out_size, void* d_ws, size_t ws_size,
                              hipStream_t stream) {
    (void)in_sizes; (void)n_in; (void)out_size; (void)ws_size;
    const float* H     = (const float*)d_in[0];  // (B, Nr, Nt, S, 2)
    const float* noise = (const float*)d_in[1];  // (B, R, Nr, Mt*S, 2)
    const float* kW    = (const float*)d_in[2];  // (Nr, Mr)
    const float* kF    = (const float*)d_in[3];  // (Nt, Mt)
    const float* kron2 = (const float*)d_in[4];  // (Nt*Nr, G^2, 2)
    float* out = (float*)d_out;
    float* ws  = (float*)d_ws;

    // Workspace layout (floats)
    float* wf   = ws;                              // 4 * 2048
    float* Qc   = ws + 8192;                       // 1024*4096*2  = 32 MB
    float* hc   = Qc + (size_t)QDIM * KDIM * 2;    // 4096*2048*2  = 64 MB
    float* effc = hc + (size_t)KDIM * YCOLS * 2;   // 1024*2048*2  = 16 MB

    float* phi_out = out;                          // (1024, 4096, 2)
    float* y_out   = out + (size_t)QDIM * GSQ * 2; // (64, 1024, 32, 2)

    phases_kernel<<<8, 256, 0, stream>>>(kW, kF, wf);
    build_q_kernel<<<dim3(KDIM / 256, QDIM), 256, 0, stream>>>(wf, Qc);
    pack_h_kernel<<<dim3(YCOLS / 256, KDIM), 256, 0, stream>>>(H, hc);
    eff_kernel<<<dim3(YCOLS / 256, QDIM), 256, 0, stream>>>(noise, wf, effc);

    // Phi = Q @ Kron2
    cgemm_wmma_kernel<0><<<dim3(GSQ / 128, QDIM / 64), 256, 0, stream>>>(
        Qc, kron2, nullptr, phi_out, GSQ);
    // y = Q @ h + n_eff  (all batches folded into N = B*S = 2048)
    cgemm_wmma_kernel<1><<<dim3(YCOLS / 128, QDIM / 64), 256, 0, stream>>>(
        Qc, hc, effc, y_out, YCOLS);
}